// CubicFeatureSampling_38397007626443
// MI455X (gfx1250) — compile-verified
//
#include <hip/hip_runtime.h>

// ---------------------------------------------------------------------------
// CubicFeatureSampling for MI455X (gfx1250, wave32)
//
// out[b][n][corner][c] = mask(corner) * feat[b][c][clamp(ix)][clamp(iy)][clamp(iz)]
//
// Pure data movement: 1 GiB NT output stream + gather of a 256 MiB feature
// tensor kept resident in the 192 MB L2 (NT stores avoid evicting it).
// z-corner pairs share one cache line -> float2 gathers halve request count.
// Point coords staged via gfx1250 async copy (global_load_async_to_lds_b32).
// ---------------------------------------------------------------------------

#define S_DIM 32
#define S3    (S_DIM * S_DIM * S_DIM)   // 32768
#define C_DIM 256
#define PTS_PER_BLOCK 4                 // 64 lanes per point, 4 channels/lane

typedef float v4f __attribute__((ext_vector_type(4)));              // 16B NT store
typedef float v2f __attribute__((ext_vector_type(2), aligned(4)));  // 8B gather (4B aligned)

__global__ __launch_bounds__(256, 8)
void cubic_feature_sampling_kernel(const float* __restrict__ ptcloud,   // [B][N][3]
                                   const float* __restrict__ feat,      // [B][C][S][S][S]
                                   float* __restrict__ out,             // [B][N][8][C]
                                   int N)
{
    __shared__ float spts[PTS_PER_BLOCK * 3];

    const int b      = blockIdx.y;
    const int n_base = blockIdx.x * PTS_PER_BLOCK;

    // ---- stage this block's point coords into LDS via ASYNCcnt path ----
    // Wave 0, lanes 0..3*n_live-1 each async-copy one float into LDS.
    const int n_live  = min(PTS_PER_BLOCK, N - n_base);
    const int n_float = 3 * n_live;
    if ((int)threadIdx.x < n_float) {
        const float* gsrc = ptcloud + ((size_t)b * N + n_base) * 3 + threadIdx.x;
        unsigned           lds_off = (unsigned)(uintptr_t)&spts[threadIdx.x];
        unsigned long long gaddr   = (unsigned long long)(uintptr_t)gsrc;
        asm volatile("global_load_async_to_lds_b32 %0, %1, off"
                     :: "v"(lds_off), "v"(gaddr) : "memory");
    }
    asm volatile("s_wait_asynccnt 0x0" ::: "memory");   // no-op for waves 1..7
    __syncthreads();                                    // publish LDS to all waves

    const int p  = threadIdx.x >> 6;        // point slot within block (0..3)
    const int cq = threadIdx.x & 63;        // channel-quad id (0..63)
    const int c0 = cq << 2;                 // first channel of this lane's quad
    const int n  = n_base + p;
    if (n >= N) return;                     // after the barrier: safe

    // --- point coordinates -> voxel lower corner (from LDS) --------------
    const float half = 0.5f * (float)S_DIM;             // 16
    const float px = spts[p * 3 + 0] * half + half;
    const float py = spts[p * 3 + 1] * half + half;
    const float pz = spts[p * 3 + 2] * half + half;
    const int lx = (int)floorf(px);
    const int ly = (int)floorf(py);
    const int lz = (int)floorf(pz);

    const float* gb = feat + (size_t)b * C_DIM * S3 + (size_t)c0 * S3;
    float*       ob = out + (((size_t)b * N + n) * 8) * C_DIM + c0;

    if ((lz >= 0) & (lz < S_DIM - 1)) {
        // ---- fast path: z and z+1 live in the same 128B z-row ----------
        // 4 xy-corners x 4 channels = 16 float2 gathers (vs 32 scalar).
#pragma unroll
        for (int j = 0; j < 4; ++j) {                   // j = dx*2 + dy
            const int dx = j >> 1, dy = j & 1;
            const int ix = lx + dx, iy = ly + dy;
            const bool v = (ix >= 0) & (ix < S_DIM) & (iy >= 0) & (iy < S_DIM);
            const int cx = min(max(ix, 0), S_DIM - 1);
            const int cy = min(max(iy, 0), S_DIM - 1);
            const int row = (cx * S_DIM + cy) * S_DIM + lz;
            const float* g = gb + row;
            __builtin_prefetch(g, 0, 3);                // global_prefetch_b8
            const v2f q0 = *(const v2f*)(g + 0 * (size_t)S3);
            const v2f q1 = *(const v2f*)(g + 1 * (size_t)S3);
            const v2f q2 = *(const v2f*)(g + 2 * (size_t)S3);
            const v2f q3 = *(const v2f*)(g + 3 * (size_t)S3);
            const float m = v ? 1.0f : 0.0f;            // mask, same as reference
            v4f r0 = { q0.x * m, q1.x * m, q2.x * m, q3.x * m };  // dz = 0
            v4f r1 = { q0.y * m, q1.y * m, q2.y * m, q3.y * m };  // dz = 1
            // corner index = dx*4 + dy*2 + dz = 2*j + dz (x-major, matches ref)
            __builtin_nontemporal_store(r0, (v4f*)(ob + (2 * j + 0) * C_DIM));
            __builtin_nontemporal_store(r1, (v4f*)(ob + (2 * j + 1) * C_DIM));
        }
    } else {
        // ---- slow path (lz == 31 edge, ~1/32 of points): scalar gathers --
#pragma unroll
        for (int corner = 0; corner < 8; ++corner) {
            const int dx = (corner >> 2) & 1;
            const int dy = (corner >> 1) & 1;
            const int dz =  corner       & 1;
            const int ix = lx + dx, iy = ly + dy, iz = lz + dz;
            const bool v = (ix >= 0) & (ix < S_DIM) &
                           (iy >= 0) & (iy < S_DIM) &
                           (iz >= 0) & (iz < S_DIM);
            const int cx = min(max(ix, 0), S_DIM - 1);
            const int cy = min(max(iy, 0), S_DIM - 1);
            const int cz = min(max(iz, 0), S_DIM - 1);
            const float* g = gb + (cx * S_DIM + cy) * S_DIM + cz;
            const float m = v ? 1.0f : 0.0f;
            v4f r = { g[0 * (size_t)S3] * m, g[1 * (size_t)S3] * m,
                      g[2 * (size_t)S3] * m, g[3 * (size_t)S3] * m };
            __builtin_nontemporal_store(r, (v4f*)(ob + corner * C_DIM));
        }
    }
}

extern "C" void kernel_launch(void* const* d_in, const int* in_sizes, int n_in,
                              void* d_out, int out_size, void* d_ws, size_t ws_size,
                              hipStream_t stream) {
    const float* ptcloud = (const float*)d_in[0];   // [B][N][3]  f32
    const float* feat    = (const float*)d_in[1];   // [B][C][S][S][S] f32
    float*       out     = (float*)d_out;           // [B][N][8][C] f32

    // Derive B and N from flat sizes (C=256, S=32 fixed by the module).
    const int B = in_sizes[1] / (C_DIM * S3);
    const int N = in_sizes[0] / (3 * B);

    dim3 grid((N + PTS_PER_BLOCK - 1) / PTS_PER_BLOCK, B);
    dim3 block(256);
    hipLaunchKernelGGL(cubic_feature_sampling_kernel, grid, block, 0, stream,
                       ptcloud, feat, out, N);
}